// AvatarGaussianEstimator_83631603188416
// MI455X (gfx1250) — compile-verified
//
#include <hip/hip_runtime.h>

typedef float    v2f   __attribute__((ext_vector_type(2)));
typedef unsigned u32x4 __attribute__((ext_vector_type(4)));
typedef int      i32x4 __attribute__((ext_vector_type(4)));
typedef int      i32x8 __attribute__((ext_vector_type(8)));

#define Bq   4
#define Cq   64
#define Hq   256
#define Wq   256
#define HWq  (Hq * Wq)
#define Nq   300000
#define Kq   16
#define NVq  6890

#define TP   64   // pixels per transpose tile
#define TROW 65   // padded LDS row stride (floats)

#if __has_builtin(__builtin_amdgcn_tensor_load_to_lds)
#define USE_TDM 1
#else
#define USE_TDM 0
#endif

// ---------------------------------------------------------------------------
// Kernel 1: NCHW -> NHWC layout transform (67 MB), feeding L2-resident NHWC.
// Load phase: one TDM tensor_load_to_lds per workgroup (2D tile: 64 channels x
// 64 contiguous pixels, row stride HW, with TDM LDS-padding producing the
// bank-conflict-free [64][65] layout). Fallback: per-lane async-to-LDS loads.
// Store phase: global_store_async_from_lds_b32 (LDS -> global, ASYNCcnt).
// ---------------------------------------------------------------------------
__global__ __launch_bounds__(256) void transpose_nchw_nhwc(
    const float* __restrict__ in, float* __restrict__ out) {
  __shared__ float tile[Cq][TROW];

  const int b    = blockIdx.x >> 10;           // 1024 pixel-tiles per batch
  const int pix0 = (blockIdx.x & 1023) << 6;   // * 64
  const int t    = threadIdx.x;

#if USE_TDM
  if (t < 32) {  // wave 0 issues the DMA (TDM ignores EXEC; one issue per wave)
    const unsigned long long gaddr =
        (unsigned long long)(size_t)in +
        ((size_t)b * Cq * HWq + (size_t)pix0) * sizeof(float);
    const unsigned lds_addr = (unsigned)(size_t)&tile[0][0];

    // D# group 0: count=1 | lds_addr | global_addr[56:0] | type=2
    u32x4 g0;
    g0.x = 1u;                                   // count=1 valid descriptor
    g0.y = lds_addr;                             // LDS byte address
    g0.z = (unsigned)(gaddr & 0xffffffffull);    // global_addr[31:0]
    g0.w = (unsigned)(gaddr >> 32) | 0x80000000u;// global_addr[56:32] | type=2

    // D# group 1:
    //  w0: data_size=2(4B)<<16 | pad_enable<<20 | pad_interval=5(64 DW)<<22
    //  tensor_dim0=65536 (bits 79:48), tensor_dim1=64 (bits 111:80)
    //  tile_dim0=64 (bits 127:112), tile_dim1=64 (bits 143:128)
    //  tensor_dim0_stride=65536 (bits 207:160)
    i32x8 g1;
    g1[0] = (int)((2u << 16) | (1u << 20) | (5u << 22)); // 0x01520000
    g1[1] = 0;                                   // abar=0, tensor_dim0 lo16=0
    g1[2] = (int)(0x1u | (64u << 16));           // tensor_dim0 hi=1, tdim1=64
    g1[3] = (int)(64u << 16);                    // tile_dim0 = 64
    g1[4] = 64;                                  // tile_dim1 = 64, tile_dim2=0
    g1[5] = (int)0x10000u;                       // stride0 = 65536
    g1[6] = 0;
    g1[7] = 0;

    const i32x4 z4 = {0, 0, 0, 0};               // dims 2..4 unused (tile=0)
#if defined(__clang_major__) && (__clang_major__ >= 23)
    // amdgpu-toolchain (clang-23+) ships the 6-arg form:
    // (uint32x4, int32x8, int32x4, int32x4, int32x8, i32 cpol)
    const i32x8 z8 = {0, 0, 0, 0, 0, 0, 0, 0};
    __builtin_amdgcn_tensor_load_to_lds(g0, g1, z4, z4, z8, 0);
#else
    // ROCm 7.2 (clang-22): 5-arg form (uint32x4, int32x8, int32x4, int32x4, i32)
    __builtin_amdgcn_tensor_load_to_lds(g0, g1, z4, z4, 0);
#endif
    __builtin_amdgcn_s_wait_tensorcnt(0);
  }
#else
  {
    const int p  = t & 63;                       // pixel within tile
    const int c0 = t >> 6;                       // channel base (0..3)
    const float* src = in + (size_t)b * Cq * HWq + pix0 + p;
    #pragma unroll
    for (int i = 0; i < 16; ++i) {
      const int c = (i << 2) + c0;
      unsigned lds_off = (unsigned)(size_t)(&tile[c][p]);
      const float* g = src + (size_t)c * HWq;
      asm volatile("global_load_async_to_lds_b32 %0, %1, off"
                   :: "v"(lds_off), "v"(g) : "memory");
    }
    asm volatile("s_wait_asynccnt 0x0" ::: "memory");
  }
#endif
  __syncthreads();

  // Store NHWC: consecutive threads -> consecutive channels (coalesced).
  // Async LDS->global stores: no VGPR staging.
  float* dst = out + ((size_t)b * HWq + pix0) * Cq;
  #pragma unroll
  for (int i = 0; i < 16; ++i) {
    const int flat = (i << 8) + t;
    const int pp   = flat >> 6;   // pixel (0..63)
    const int cc   = flat & 63;   // channel
    unsigned lds_off = (unsigned)(size_t)(&tile[cc][pp]);
    float* g = dst + (size_t)pp * Cq + cc;
    asm volatile("global_store_async_from_lds_b32 %0, %1, off"
                 :: "v"(g), "v"(lds_off) : "memory");
  }
  asm volatile("s_wait_asynccnt 0x0" ::: "memory");
}

// ---------------------------------------------------------------------------
// Kernel 2: per-point bilinear sample from NHWC. One wave32 per point,
// each lane owns 2 channels -> each corner = one coalesced 256B wave load.
// ---------------------------------------------------------------------------
__device__ __forceinline__ v2f corner_nhwc(const float* __restrict__ fmt,
                                           unsigned b, int xi, int yi, int lane) {
  const bool valid = (xi >= 0) & (xi <= Wq - 1) & (yi >= 0) & (yi <= Hq - 1);
  const int xc = min(max(xi, 0), Wq - 1);
  const int yc = min(max(yi, 0), Hq - 1);
  const float* pb = fmt + (((size_t)b * Hq + yc) * Wq + xc) * Cq + lane * 2;
  v2f v = *(const v2f*)pb;
  const float m = valid ? 1.0f : 0.0f;
  v.x *= m; v.y *= m;
  return v;
}

__global__ __launch_bounds__(256) void sample_nhwc(
    const float* __restrict__ fmt, const float* __restrict__ verts,
    const float* __restrict__ bary, const int* __restrict__ parents,
    float* __restrict__ out) {
  const int t    = threadIdx.x;
  const int lane = t & 31;
  // 8 points per block; 37500 blocks per batch (300000/8): no batch straddle,
  // so only a cheap 32-bit const-divide is needed (no 64-bit div chain).
  const unsigned bidx = blockIdx.x;
  const unsigned b    = bidx / 37500u;
  const unsigned n    = (bidx - b * 37500u) * 8u + (unsigned)(t >> 5);

  // barycentric center (lane-uniform per wave -> coalesced scalar-ish loads)
  const unsigned kb = n & (Kq - 1);
  const float w0 = bary[kb * 3 + 0], w1 = bary[kb * 3 + 1], w2 = bary[kb * 3 + 2];
  const int p0 = parents[n * 3 + 0], p1 = parents[n * 3 + 1], p2 = parents[n * 3 + 2];
  const float* vb = verts + (size_t)b * (NVq * 2);
  float px = vb[2 * p0]     * w0 + vb[2 * p1]     * w1 + vb[2 * p2]     * w2;
  float py = vb[2 * p0 + 1] * w0 + vb[2 * p1 + 1] * w1 + vb[2 * p2 + 1] * w2;

  // reference's normalize/denormalize round-trip (identity up to f32 rounding)
  const float gx = px / (float)(Wq - 1) * 2.0f - 1.0f;
  const float gy = py / (float)(Hq - 1) * 2.0f - 1.0f;
  px = (gx + 1.0f) * 0.5f * (float)(Wq - 1);
  py = (gy + 1.0f) * 0.5f * (float)(Hq - 1);

  const float x0f = floorf(px), y0f = floorf(py);
  const float wx = px - x0f, wy = py - y0f;
  const int x0 = (int)x0f, y0 = (int)y0f;

  const v2f v00 = corner_nhwc(fmt, b, x0,     y0,     lane);
  const v2f v01 = corner_nhwc(fmt, b, x0 + 1, y0,     lane);
  const v2f v10 = corner_nhwc(fmt, b, x0,     y0 + 1, lane);
  const v2f v11 = corner_nhwc(fmt, b, x0 + 1, y0 + 1, lane);

  const float w00 = (1.0f - wx) * (1.0f - wy);
  const float w01 = wx * (1.0f - wy);
  const float w10 = (1.0f - wx) * wy;
  const float w11 = wx * wy;

  v2f acc;
  acc.x = v00.x * w00 + v01.x * w01 + v10.x * w10 + v11.x * w11;
  acc.y = v00.y * w00 + v01.y * w01 + v10.y * w10 + v11.y * w11;

  // NT store: don't let the 307MB output stream evict the L2-resident fmap
  float* po = out + ((size_t)b * Nq + n) * Cq + lane * 2;
  __builtin_nontemporal_store(acc, (v2f*)po);
}

// ---------------------------------------------------------------------------
// Fallback: direct NCHW sampling (if workspace too small for the transpose).
// ---------------------------------------------------------------------------
__device__ __forceinline__ v2f corner_nchw(const float* __restrict__ fm,
                                           unsigned b, int xi, int yi, int c) {
  const bool valid = (xi >= 0) & (xi <= Wq - 1) & (yi >= 0) & (yi <= Hq - 1);
  const int xc = min(max(xi, 0), Wq - 1);
  const int yc = min(max(yi, 0), Hq - 1);
  const float* pb = fm + ((size_t)(b * Cq + c) * Hq + yc) * Wq + xc;
  const float m = valid ? 1.0f : 0.0f;
  v2f v;
  v.x = pb[0]   * m;
  v.y = pb[HWq] * m;  // channel c+1
  return v;
}

__global__ __launch_bounds__(256) void sample_nchw(
    const float* __restrict__ fm, const float* __restrict__ verts,
    const float* __restrict__ bary, const int* __restrict__ parents,
    float* __restrict__ out) {
  const int t    = threadIdx.x;
  const int lane = t & 31;
  const unsigned bidx = blockIdx.x;
  const unsigned b    = bidx / 37500u;
  const unsigned n    = (bidx - b * 37500u) * 8u + (unsigned)(t >> 5);

  const unsigned kb = n & (Kq - 1);
  const float w0 = bary[kb * 3 + 0], w1 = bary[kb * 3 + 1], w2 = bary[kb * 3 + 2];
  const int p0 = parents[n * 3 + 0], p1 = parents[n * 3 + 1], p2 = parents[n * 3 + 2];
  const float* vb = verts + (size_t)b * (NVq * 2);
  float px = vb[2 * p0]     * w0 + vb[2 * p1]     * w1 + vb[2 * p2]     * w2;
  float py = vb[2 * p0 + 1] * w0 + vb[2 * p1 + 1] * w1 + vb[2 * p2 + 1] * w2;

  const float gx = px / (float)(Wq - 1) * 2.0f - 1.0f;
  const float gy = py / (float)(Hq - 1) * 2.0f - 1.0f;
  px = (gx + 1.0f) * 0.5f * (float)(Wq - 1);
  py = (gy + 1.0f) * 0.5f * (float)(Hq - 1);

  const float x0f = floorf(px), y0f = floorf(py);
  const float wx = px - x0f, wy = py - y0f;
  const int x0 = (int)x0f, y0 = (int)y0f;
  const int c = lane * 2;

  const v2f v00 = corner_nchw(fm, b, x0,     y0,     c);
  const v2f v01 = corner_nchw(fm, b, x0 + 1, y0,     c);
  const v2f v10 = corner_nchw(fm, b, x0,     y0 + 1, c);
  const v2f v11 = corner_nchw(fm, b, x0 + 1, y0 + 1, c);

  const float w00 = (1.0f - wx) * (1.0f - wy);
  const float w01 = wx * (1.0f - wy);
  const float w10 = (1.0f - wx) * wy;
  const float w11 = wx * wy;

  v2f acc;
  acc.x = v00.x * w00 + v01.x * w01 + v10.x * w10 + v11.x * w11;
  acc.y = v00.y * w00 + v01.y * w01 + v10.y * w10 + v11.y * w11;

  float* po = out + ((size_t)b * Nq + n) * Cq + c;
  __builtin_nontemporal_store(acc, (v2f*)po);
}

// ---------------------------------------------------------------------------
extern "C" void kernel_launch(void* const* d_in, const int* in_sizes, int n_in,
                              void* d_out, int out_size, void* d_ws, size_t ws_size,
                              hipStream_t stream) {
  (void)in_sizes; (void)n_in; (void)out_size;
  const float* fmap    = (const float*)d_in[0];
  const float* verts   = (const float*)d_in[1];
  const float* bary    = (const float*)d_in[2];
  const int*   parents = (const int*)d_in[3];
  float* out = (float*)d_out;

  const size_t need = (size_t)Bq * Cq * HWq * sizeof(float);  // 67 MB NHWC
  const int blocks = (int)(((long long)Bq * Nq + 7) / 8);     // 150000

  if (ws_size >= need) {
    float* fmt = (float*)d_ws;
    transpose_nchw_nhwc<<<Bq * (HWq / TP), 256, 0, stream>>>(fmap, fmt);
    sample_nhwc<<<blocks, 256, 0, stream>>>(fmt, verts, bary, parents, out);
  } else {
    sample_nchw<<<blocks, 256, 0, stream>>>(fmap, verts, bary, parents, out);
  }
}